// DPSR2_30210799960071
// MI455X (gfx1250) — compile-verified
//
#include <hip/hip_runtime.h>
#include <math.h>

// ---------------------------------------------------------------------------
// DPSR forward for MI455X (gfx1250, wave32).
//
// Pipeline (all f32):
//   1) zero + trilinear scatter of normals into G0[b][c][x][y][z]   (atomics)
//   2) rfftn as DFT-GEMMs against 128x128 cos/sin tables, one axis at a time,
//      executed with V_WMMA_F32_16X16X4_F32 (f32 matrix pipe keeps precision;
//      whole working set is L2-resident: 100.7MB grid < 192MB L2)
//   3) spectral divide (replicating the np.meshgrid 'xy' channel/k pairing)
//      with the irfft Hermitian weights (x2 for kz = 1..63) folded in
//   4) irfftn as three more DFT-GEMM stages (alpha = 1/128 each)
//   5) trilinear gather at V, per-batch mean + normalization
//
// v2: 16x32 output tile per wave (2 WMMAs per A fragment), W tile transposed
//     in LDS so both A and B fragments are single ds_load_b64, unroll-8 K loop
//     so LDS latency pipelines behind WMMAs, float4 global loads for the
//     contiguous-stride stages.
// ---------------------------------------------------------------------------

typedef __attribute__((ext_vector_type(2))) float v2f;
typedef __attribute__((ext_vector_type(8))) float v8f;

#define RES 128
#define NB 4
#define NCH 3
#define NPTS 65536
#define GRID3 2097152LL          // 128^3
#define TWO_PI_128 0.04908738521234052f

// ---- workspace layout (float offsets) -------------------------------------
#define OFF_G0   0LL             // 25165824 floats  (BufA; later Pre/Pim, T2)
#define OFF_S1RE 25165824LL      // 12779520 floats  (BufB; later S3re, phi)
#define OFF_S1IM 37945344LL      // 12779520
#define OFF_S2RE 50724864LL      // 12779520 floats  (BufC; later T1)
#define OFF_S2IM 63504384LL      // 12779520
#define OFF_TAB  76283904LL      // 3 * 16384 (COS | SIN | NSIN)
#define OFF_ACC  76333056LL      // 12 floats: sums[4], mean[4], scale[4]
#define HALF_CPLX 4259840LL      // 4*128*128*65

// ---------------------------------------------------------------------------
__global__ void zero_f32_kernel(float* __restrict__ p, long long n) {
  long long i = (long long)blockIdx.x * blockDim.x + threadIdx.x;
  long long stride = (long long)gridDim.x * blockDim.x;
  for (; i < n; i += stride) p[i] = 0.0f;
}

__global__ void init_tables_kernel(float* __restrict__ tab) {
  int i = blockIdx.x * 256 + threadIdx.x;
  if (i >= 16384) return;
  int r = i >> 7, c = i & 127;
  int m = (r * c) & 127;                 // arg reduction: cos/sin(2*pi*m/128)
  float s, co;
  __sincosf(TWO_PI_128 * (float)m, &s, &co);
  tab[i]           = co;                 // COS
  tab[16384 + i]   = s;                  // SIN
  tab[32768 + i]   = -s;                 // NSIN
}

// ---- trilinear scatter -----------------------------------------------------
__global__ void rasterize_kernel(const float* __restrict__ V,
                                 const float* __restrict__ N,
                                 float* __restrict__ G) {
  int i = blockIdx.x * blockDim.x + threadIdx.x;
  if (i >= NB * NPTS) return;
  int b = i >> 16;
  const float* v  = V + (size_t)i * 3;
  const float* nn = N + (size_t)i * 3;
  int i0[3], i1[3]; float fr[3];
#pragma unroll
  for (int d = 0; d < 3; ++d) {
    float x  = v[d] * 128.0f;
    float f0 = floorf(x);
    i0[d] = (int)f0;
    i1[d] = ((int)ceilf(x)) & 127;       // mod 128 (ref: mod(ceil(x), res))
    fr[d] = x - f0;
  }
  float nv0 = nn[0], nv1 = nn[1], nv2 = nn[2];
#pragma unroll
  for (int corner = 0; corner < 8; ++corner) {
    int cx = (corner >> 2) & 1, cy = (corner >> 1) & 1, cz = corner & 1;
    int ix = cx ? i1[0] : i0[0];
    int iy = cy ? i1[1] : i0[1];
    int iz = cz ? i1[2] : i0[2];
    float w = (cx ? fr[0] : 1.0f - fr[0]) *
              (cy ? fr[1] : 1.0f - fr[1]) *
              (cz ? fr[2] : 1.0f - fr[2]);
    long long cell = ((long long)ix * 128 + iy) * 128 + iz;
    long long base = ((long long)b * 3) * GRID3 + cell;
    atomicAdd(&G[base],             w * nv0);
    atomicAdd(&G[base + GRID3],     w * nv1);
    atomicAdd(&G[base + 2 * GRID3], w * nv2);
  }
}

// ---- generic strided DFT-GEMM with f32 WMMA -------------------------------
// C[line, col] = alpha * ( sum_k A1[line,k]*W1[k,col] + A2[line,k]*W2[k,col] )
// line -> (outer = line/inner_count, inner = line%inner_count)
// A addr = outer*a_outer + inner*a_inner + k*a_col       (floats)
// C addr = outer*c_outer + inner*c_inner + col*c_col
// W tables are 128x128 row-major.
// One wave computes a 16x32 output tile (two 16x16 WMMA accumulators sharing
// each A fragment); 4 waves/block share a transposed 32x132 W tile in LDS.
// Both A and B fragments are contiguous pairs -> single ds_load_b64 each;
// row pad 132 makes bank = (4*row + k) % 64 -> conflict-free.
__launch_bounds__(128)
__global__ void dft_gemm_kernel(const float* __restrict__ A1,
                                const float* __restrict__ A2,
                                const float* __restrict__ W1,
                                const float* __restrict__ W2,
                                float* __restrict__ C,
                                int K, int Ncols, long long inner_count,
                                long long a_inner, long long a_outer, long long a_col,
                                long long c_inner, long long c_outer, long long c_col,
                                float alpha, int n_pairs) {
  __shared__ float Wl[32][132];            // transposed: Wl[n][k] = W[k][n0+n]
  __shared__ float Al[4][16][132];

  const int tid  = threadIdx.x;
  const int wv   = tid >> 5;
  const int lane = tid & 31;
  const int npair = blockIdx.x % n_pairs;
  const int mb4   = blockIdx.x / n_pairs;
  const int tile_m = mb4 * 4 + wv;
  const int n0   = npair * 32;
  const int Kpad = (K + 3) & ~3;
  const bool useA2 = (A2 != nullptr);
  const bool vecA  = (a_col == 1) && (Kpad == 128);

  // ---- W1 tile (cooperative, transposed; zero-padded rows/cols) ----
  for (int idx = tid; idx < 32 * 128; idx += 128) {
    int nn = idx & 31, kk = idx >> 5;
    int col = n0 + nn;
    Wl[nn][kk] = ((kk < K) && (col < Ncols)) ? W1[kk * 128 + col] : 0.0f;
  }

  // ---- A1 tile (per wave): 2 lanes per row ----
  {
    const int r = lane >> 1;
    const long long line = (long long)tile_m * 16 + r;
    const long long base = (line / inner_count) * a_outer +
                           (line % inner_count) * a_inner;
    if (vecA) {                                  // contiguous rows, 16B aligned
      const float4* src = (const float4*)(A1 + base);
      float4* dst = (float4*)&Al[wv][r][0];
      const int h4 = (lane & 1) * 16;
#pragma unroll
      for (int kk = 0; kk < 16; ++kk) dst[h4 + kk] = src[h4 + kk];
    } else {
      const int h = (lane & 1) * (Kpad >> 1);
      for (int kk = 0; kk < (Kpad >> 1); ++kk) {
        int k = h + kk;
        Al[wv][r][k] = (k < K) ? A1[base + (long long)k * a_col] : 0.0f;
      }
    }
  }
  __syncthreads();

  v8f acc0 = {0, 0, 0, 0, 0, 0, 0, 0};
  v8f acc1 = {0, 0, 0, 0, 0, 0, 0, 0};
  const int m  = lane & 15;                // A-row per lane
  const int nl = lane & 15;                // B/C-col per lane
  const int ko = (lane >> 4) << 1;         // K sub-offset: lanes16-31 -> K+2

#pragma unroll 8
  for (int k0 = 0; k0 < Kpad; k0 += 4) {
    v2f a, b0, b1;
    a.x  = Al[wv][m][k0 + ko];
    a.y  = Al[wv][m][k0 + ko + 1];
    b0.x = Wl[nl][k0 + ko];
    b0.y = Wl[nl][k0 + ko + 1];
    b1.x = Wl[16 + nl][k0 + ko];
    b1.y = Wl[16 + nl][k0 + ko + 1];
    acc0 = __builtin_amdgcn_wmma_f32_16x16x4_f32(false, a, false, b0,
                                                 (short)0, acc0, false, false);
    acc1 = __builtin_amdgcn_wmma_f32_16x16x4_f32(false, a, false, b1,
                                                 (short)0, acc1, false, false);
  }

  if (useA2) {                             // uniform branch: EXEC stays full
    __syncthreads();                       // everyone done reading Wl/Al
    for (int idx = tid; idx < 32 * 128; idx += 128) {
      int nn = idx & 31, kk = idx >> 5;
      int col = n0 + nn;
      Wl[nn][kk] = ((kk < K) && (col < Ncols)) ? W2[kk * 128 + col] : 0.0f;
    }
    {
      const int r = lane >> 1;
      const long long line = (long long)tile_m * 16 + r;
      const long long base = (line / inner_count) * a_outer +
                             (line % inner_count) * a_inner;
      if (vecA) {
        const float4* src = (const float4*)(A2 + base);
        float4* dst = (float4*)&Al[wv][r][0];
        const int h4 = (lane & 1) * 16;
#pragma unroll
        for (int kk = 0; kk < 16; ++kk) dst[h4 + kk] = src[h4 + kk];
      } else {
        const int h = (lane & 1) * (Kpad >> 1);
        for (int kk = 0; kk < (Kpad >> 1); ++kk) {
          int k = h + kk;
          Al[wv][r][k] = (k < K) ? A2[base + (long long)k * a_col] : 0.0f;
        }
      }
    }
    __syncthreads();
#pragma unroll 8
    for (int k0 = 0; k0 < Kpad; k0 += 4) {
      v2f a, b0, b1;
      a.x  = Al[wv][m][k0 + ko];
      a.y  = Al[wv][m][k0 + ko + 1];
      b0.x = Wl[nl][k0 + ko];
      b0.y = Wl[nl][k0 + ko + 1];
      b1.x = Wl[16 + nl][k0 + ko];
      b1.y = Wl[16 + nl][k0 + ko + 1];
      acc0 = __builtin_amdgcn_wmma_f32_16x16x4_f32(false, a, false, b0,
                                                   (short)0, acc0, false, false);
      acc1 = __builtin_amdgcn_wmma_f32_16x16x4_f32(false, a, false, b1,
                                                   (short)0, acc1, false, false);
    }
  }

  // store: C/D layout -> VGPR j holds M=j (lanes 0-15) and M=j+8 (lanes 16-31)
#pragma unroll
  for (int t = 0; t < 2; ++t) {
    const int col = n0 + t * 16 + nl;
    if (col < Ncols) {
#pragma unroll
      for (int j = 0; j < 8; ++j) {
        int mr = (lane < 16) ? j : (j + 8);
        long long line = (long long)tile_m * 16 + mr;
        long long cb = (line / inner_count) * c_outer +
                       (line % inner_count) * c_inner;
        C[cb + (long long)col * c_col] = (t ? acc1[j] : acc0[j]) * alpha;
      }
    }
  }
}

// ---- spectral divide -------------------------------------------------------
__global__ void spectral_kernel(const float* __restrict__ S3re,
                                const float* __restrict__ S3im,
                                float* __restrict__ Pre,
                                float* __restrict__ Pim) {
  long long i = (long long)blockIdx.x * 256 + threadIdx.x;
  if (i >= 4LL * 128 * 128 * 65) return;
  int kz = (int)(i % 65);
  long long t = i / 65;
  int ky = (int)(t % 128); t /= 128;
  int kx = (int)(t % 128);
  int b  = (int)(t / 128);
  float sy, cy; __sincosf(TWO_PI_128 * (float)ky, &sy, &cy);
  float sx, cx; __sincosf(TWO_PI_128 * (float)kx, &sx, &cx);
  float sz, cz; __sincosf(TWO_PI_128 * (float)kz, &sz, &cz);
  // np.meshgrid default 'xy': channel0 pairs with ky, channel1 with kx.
  float s0 = 128.0f * sy, s1 = 128.0f * sx, s2 = 128.0f * sz;
  float L = 2.0f * 16384.0f * (cx + cy + cz - 3.0f) + 1e-6f;
  const long long cs = 1064960LL;          // 128*128*65 per (b,c)
  long long base = (long long)b * 3 * cs + (long long)kx * 8320 +
                   (long long)ky * 65 + kz;
  float re0 = S3re[base],          im0 = S3im[base];
  float re1 = S3re[base + cs],     im1 = S3im[base + cs];
  float re2 = S3re[base + 2 * cs], im2 = S3im[base + 2 * cs];
  float invL  = 1.0f / L;
  float pr = -(s0 * im0 + s1 * im1 + s2 * im2) * invL;
  float pi =  (s0 * re0 + s1 * re1 + s2 * re2) * invL;
  if (kx == 0 && ky == 0 && kz == 0) { pr = 0.0f; pi = 0.0f; }
  float coef = (kz == 0 || kz == 64) ? 1.0f : 2.0f;  // irfft Hermitian weight
  long long po = (long long)b * cs + (long long)kx * 8320 +
                 (long long)ky * 65 + kz;
  Pre[po] = pr * coef;
  Pim[po] = pi * coef;
}

// ---- trilinear gather + per-batch mean ------------------------------------
__global__ void interp_kernel(const float* __restrict__ V,
                              const float* __restrict__ phi,
                              float* __restrict__ acc) {
  __shared__ float red[256];
  int i = blockIdx.x * 256 + threadIdx.x;   // blocks never straddle batches
  int b = i >> 16;
  const float* v = V + (size_t)i * 3;
  int i0[3], i1[3]; float fr[3];
#pragma unroll
  for (int d = 0; d < 3; ++d) {
    float x  = v[d] * 128.0f;
    float f0 = floorf(x);
    i0[d] = (int)f0;
    i1[d] = ((int)ceilf(x)) & 127;
    fr[d] = x - f0;
  }
  const float* pb = phi + (long long)b * GRID3;
  float fv = 0.0f;
#pragma unroll
  for (int corner = 0; corner < 8; ++corner) {
    int cx = (corner >> 2) & 1, cy = (corner >> 1) & 1, cz = corner & 1;
    int ix = cx ? i1[0] : i0[0];
    int iy = cy ? i1[1] : i0[1];
    int iz = cz ? i1[2] : i0[2];
    float w = (cx ? fr[0] : 1.0f - fr[0]) *
              (cy ? fr[1] : 1.0f - fr[1]) *
              (cz ? fr[2] : 1.0f - fr[2]);
    fv += w * pb[((long long)ix * 128 + iy) * 128 + iz];
  }
  red[threadIdx.x] = fv;
  __syncthreads();
  for (int s = 128; s > 0; s >>= 1) {
    if (threadIdx.x < s) red[threadIdx.x] += red[threadIdx.x + s];
    __syncthreads();
  }
  if (threadIdx.x == 0) atomicAdd(&acc[b], red[0]);
}

__global__ void finalize_kernel(const float* __restrict__ phi,
                                float* __restrict__ acc) {
  int b = threadIdx.x;
  if (b >= NB) return;
  float meanv = acc[b] * (1.0f / 65536.0f);
  float fv0 = phi[(long long)b * GRID3] - meanv;
  acc[4 + b] = meanv;
  acc[8 + b] = -0.5f / fabsf(fv0);
}

__global__ void final_out_kernel(const float* __restrict__ phi,
                                 const float* __restrict__ acc,
                                 float* __restrict__ out) {
  long long i = (long long)blockIdx.x * 256 + threadIdx.x;
  if (i >= (long long)NB * GRID3) return;
  int b = (int)(i >> 21);
  out[i] = (phi[i] - acc[4 + b]) * acc[8 + b];
}

// ---------------------------------------------------------------------------
static void launch_dft(hipStream_t s, const float* A1, const float* A2,
                       const float* W1, const float* W2, float* C,
                       int K, int Ncols, long long M, long long inner,
                       long long ai, long long ao, long long ac,
                       long long ci, long long co, long long cc, float alpha) {
  int m_tiles = (int)(M / 16);              // all stage Ms divisible by 64
  int n_pairs = (Ncols + 31) / 32;
  dim3 grid((m_tiles / 4) * n_pairs);
  dft_gemm_kernel<<<grid, dim3(128), 0, s>>>(A1, A2, W1, W2, C, K, Ncols,
                                             inner, ai, ao, ac, ci, co, cc,
                                             alpha, n_pairs);
}

extern "C" void kernel_launch(void* const* d_in, const int* in_sizes, int n_in,
                              void* d_out, int out_size, void* d_ws,
                              size_t ws_size, hipStream_t stream) {
  const float* V = (const float*)d_in[0];   // (4, 65536, 3) f32
  const float* N = (const float*)d_in[1];   // (4, 65536, 3) f32
  float* out = (float*)d_out;               // (4, 128,128,128) f32
  float* ws  = (float*)d_ws;

  float* G0   = ws + OFF_G0;
  float* S1re = ws + OFF_S1RE;
  float* S1im = ws + OFF_S1IM;
  float* S2re = ws + OFF_S2RE;
  float* S2im = ws + OFF_S2IM;
  float* S3re = S1re;                        // BufB reuse
  float* S3im = S1im;
  float* Pre  = ws + 0;                      // BufA reuse
  float* Pim  = ws + HALF_CPLX;
  float* T1re = S2re;                        // BufC reuse
  float* T1im = S2re + HALF_CPLX;
  float* T2re = ws + 0;                      // BufA reuse
  float* T2im = ws + HALF_CPLX;
  float* phi  = S1re;                        // BufB reuse (8.39M <= 12.78M)
  float* COS  = ws + OFF_TAB;
  float* SIN  = COS + 16384;
  float* NSIN = COS + 32768;
  float* acc  = ws + OFF_ACC;

  const float inv128 = 1.0f / 128.0f;

  // 0) tables + zeroing (required every call: rasterize/interp accumulate)
  init_tables_kernel<<<64, 256, 0, stream>>>(COS);
  zero_f32_kernel<<<4096, 256, 0, stream>>>(G0, (long long)NB * NCH * GRID3);
  zero_f32_kernel<<<1, 32, 0, stream>>>(acc, 12);

  // 1) trilinear scatter
  rasterize_kernel<<<(NB * NPTS) / 256, 256, 0, stream>>>(V, N, G0);

  // 2) forward z (real -> 65 complex):  lines (b,c,x,y), contiguous
  launch_dft(stream, G0, nullptr, COS, COS, S1re, 128, 65, 196608, 196608,
             128, 0, 1, 65, 0, 1, 1.0f);
  launch_dft(stream, G0, nullptr, NSIN, COS, S1im, 128, 65, 196608, 196608,
             128, 0, 1, 65, 0, 1, 1.0f);

  // 3) forward y (complex, exp(-i)):  lines (b,c,x,kz)
  launch_dft(stream, S1re, S1im, COS, SIN, S2re, 128, 128, 99840, 65,
             1, 8320, 65, 1, 8320, 65, 1.0f);
  launch_dft(stream, S1im, S1re, COS, NSIN, S2im, 128, 128, 99840, 65,
             1, 8320, 65, 1, 8320, 65, 1.0f);

  // 4) forward x (complex):  lines (b,c,ky,kz)
  launch_dft(stream, S2re, S2im, COS, SIN, S3re, 128, 128, 99840, 8320,
             1, 1064960, 8320, 1, 1064960, 8320, 1.0f);
  launch_dft(stream, S2im, S2re, COS, NSIN, S3im, 128, 128, 99840, 8320,
             1, 1064960, 8320, 1, 1064960, 8320, 1.0f);

  // 5) spectral divide (+ fold irfft Hermitian weights)
  spectral_kernel<<<(int)((4LL * 128 * 128 * 65 + 255) / 256), 256, 0,
                   stream>>>(S3re, S3im, Pre, Pim);

  // 6) inverse x (complex, exp(+i), 1/128):  lines (b,ky,kz)
  launch_dft(stream, Pre, Pim, COS, NSIN, T1re, 128, 128, 33280, 8320,
             1, 1064960, 8320, 1, 1064960, 8320, inv128);
  launch_dft(stream, Pre, Pim, SIN, COS, T1im, 128, 128, 33280, 8320,
             1, 1064960, 8320, 1, 1064960, 8320, inv128);

  // 7) inverse y:  lines (b,x,kz)
  launch_dft(stream, T1re, T1im, COS, NSIN, T2re, 128, 128, 33280, 65,
             1, 8320, 65, 1, 8320, 65, inv128);
  launch_dft(stream, T1re, T1im, SIN, COS, T2im, 128, 128, 33280, 65,
             1, 8320, 65, 1, 8320, 65, inv128);

  // 8) inverse z (65 complex -> 128 real):  lines (b,x,y), contiguous
  launch_dft(stream, T2re, T2im, COS, NSIN, phi, 65, 128, 65536, 65536,
             65, 0, 1, 128, 0, 1, inv128);

  // 9) gather, mean, normalize
  interp_kernel<<<(NB * NPTS) / 256, 256, 0, stream>>>(V, phi, acc);
  finalize_kernel<<<1, 32, 0, stream>>>(phi, acc);
  final_out_kernel<<<(int)(((long long)NB * GRID3 + 255) / 256), 256, 0,
                    stream>>>(phi, acc, out);
}